// GCNEncoder_68436008894840
// MI455X (gfx1250) — compile-verified
//
#include <hip/hip_runtime.h>

typedef __attribute__((ext_vector_type(2))) float v2f;
typedef __attribute__((ext_vector_type(8))) float v8f;

#define IN_CH  256
#define HID_CH 128
#define OUT_CH 64

// ---------------- utility fills ----------------
__global__ void fill_kernel(float* __restrict__ p, float v, long long n) {
  long long i = (long long)blockIdx.x * blockDim.x + threadIdx.x;
  long long stride = (long long)gridDim.x * blockDim.x;
  for (; i < n; i += stride) p[i] = v;
}

// ---------------- degree / norm ----------------
__global__ void deg_kernel(const int* __restrict__ dst, const float* __restrict__ ew,
                           float* __restrict__ deg, int E) {
  int e = blockIdx.x * blockDim.x + threadIdx.x;
  if (e < E) atomicAdd(&deg[dst[e]], ew[e]);
}

__global__ void dinv_kernel(float* __restrict__ deg, int n) {
  int i = blockIdx.x * blockDim.x + threadIdx.x;
  if (i < n) {
    float d = deg[i];
    deg[i] = (d > 0.0f) ? rsqrtf(fmaxf(d, 1e-12f)) : 0.0f;
  }
}

// ---------------- GEMM1: Hlin[N,128] = X[N,256] @ W1[256,128] (f32 WMMA) -------
// One wave computes a 16x128 output stripe: 8 accumulators of 16x16.
// A-frag layout (32-bit A 16x4): lanes 0-15 hold K=0,1; lanes 16-31 hold K=2,3.
// B-frag layout (32-bit B 4x16): N across lanes; VGPR0/1 = K rows (2*half, 2*half+1).
__global__ void gemm1_kernel(const float* __restrict__ X, const float* __restrict__ W,
                             float* __restrict__ H, int mtiles) {
  int wave  = threadIdx.x >> 5;
  int mtile = blockIdx.x * (blockDim.x >> 5) + wave;
  if (mtile >= mtiles) return;                    // wave-uniform: EXEC stays all-1s
  int lane = threadIdx.x & 31;
  int half = lane >> 4;
  int lm   = lane & 15;
  long long m0 = (long long)mtile * 16;

  v8f acc[8];
#pragma unroll
  for (int t = 0; t < 8; ++t) acc[t] = 0.0f;

  const float* xrow = X + (m0 + lm) * IN_CH;
  for (int k0 = 0; k0 < IN_CH; k0 += 4) {
    v2f a;
    a.x = xrow[k0 + 2 * half];
    a.y = xrow[k0 + 2 * half + 1];
#pragma unroll
    for (int t = 0; t < 8; ++t) {
      v2f b;
      b.x = W[(k0 + 2 * half)     * HID_CH + t * 16 + lm];
      b.y = W[(k0 + 2 * half + 1) * HID_CH + t * 16 + lm];
      acc[t] = __builtin_amdgcn_wmma_f32_16x16x4_f32(
          /*neg_a=*/false, a, /*neg_b=*/false, b,
          /*c_mod=*/(short)0, acc[t], /*reuse_a=*/false, /*reuse_b=*/false);
    }
  }
  // C/D layout: VGPR r -> row (r + 8*half), col = lm (within tile t)
#pragma unroll
  for (int t = 0; t < 8; ++t)
#pragma unroll
    for (int r = 0; r < 8; ++r)
      H[(m0 + r + 8 * half) * HID_CH + t * 16 + lm] = acc[t][r];
}

// ---------------- GEMM2: O[N,128] = H[N,128] @ [Wmu | Wlv] ----------------
__global__ void gemm2_kernel(const float* __restrict__ H, const float* __restrict__ Wmu,
                             const float* __restrict__ Wlv, float* __restrict__ O,
                             int mtiles) {
  int wave  = threadIdx.x >> 5;
  int mtile = blockIdx.x * (blockDim.x >> 5) + wave;
  if (mtile >= mtiles) return;
  int lane = threadIdx.x & 31;
  int half = lane >> 4;
  int lm   = lane & 15;
  long long m0 = (long long)mtile * 16;

  v8f acc[8];
#pragma unroll
  for (int t = 0; t < 8; ++t) acc[t] = 0.0f;

  const float* hrow = H + (m0 + lm) * HID_CH;
  for (int k0 = 0; k0 < HID_CH; k0 += 4) {
    v2f a;
    a.x = hrow[k0 + 2 * half];
    a.y = hrow[k0 + 2 * half + 1];
#pragma unroll
    for (int t = 0; t < 8; ++t) {
      const float* Wp = (t < 4) ? Wmu : Wlv;      // cols 0-63 = mu, 64-127 = logvar
      int col = (t & 3) * 16 + lm;
      v2f b;
      b.x = Wp[(k0 + 2 * half)     * OUT_CH + col];
      b.y = Wp[(k0 + 2 * half + 1) * OUT_CH + col];
      acc[t] = __builtin_amdgcn_wmma_f32_16x16x4_f32(
          false, a, false, b, (short)0, acc[t], false, false);
    }
  }
#pragma unroll
  for (int t = 0; t < 8; ++t)
#pragma unroll
    for (int r = 0; r < 8; ++r)
      O[(m0 + r + 8 * half) * HID_CH + t * 16 + lm] = acc[t][r];
}

// ---------------- conv1 scatter: one wave per edge, float4 per lane ----------
__global__ void scatter1_kernel(const int* __restrict__ src, const int* __restrict__ dst,
                                const float* __restrict__ ew, const float* __restrict__ dis,
                                const float* __restrict__ Hlin, float* __restrict__ Hout,
                                int E, int Nn) {
  int eid = blockIdx.x * (blockDim.x >> 5) + (threadIdx.x >> 5);
  if (eid >= E + Nn) return;
  int j = (threadIdx.x & 31) * 4;                 // feature offset 0..124
  int s, d; float w;
  if (eid < E) { s = src[eid]; d = dst[eid]; w = ew[eid]; }
  else         { s = eid - E;  d = s;        w = 1.0f;    }    // self-loop
  float nrm = dis[s] * w * dis[d];
  float4 v = *(const float4*)(Hlin + (size_t)s * HID_CH + j);
  float* o = Hout + (size_t)d * HID_CH + j;
  atomicAdd(o + 0, v.x * nrm);
  atomicAdd(o + 1, v.y * nrm);
  atomicAdd(o + 2, v.z * nrm);
  atomicAdd(o + 3, v.w * nrm);
}

__global__ void bias_relu_kernel(float* __restrict__ H, const float* __restrict__ b,
                                 long long total) {
  long long i = (long long)blockIdx.x * blockDim.x + threadIdx.x;
  long long stride = (long long)gridDim.x * blockDim.x;
  for (; i < total; i += stride) {
    int col = (int)(i & (HID_CH - 1));
    H[i] = fmaxf(H[i] + b[col], 0.0f);
  }
}

// ---------------- conv2 scatter: mu/lv fused, split write into d_out ---------
__global__ void scatter2_kernel(const int* __restrict__ src, const int* __restrict__ dst,
                                const float* __restrict__ ew, const float* __restrict__ dis,
                                const float* __restrict__ MLV, float* __restrict__ mu,
                                float* __restrict__ lv, int E, int Nn) {
  int eid = blockIdx.x * (blockDim.x >> 5) + (threadIdx.x >> 5);
  if (eid >= E + Nn) return;
  int j = (threadIdx.x & 31) * 4;
  int s, d; float w;
  if (eid < E) { s = src[eid]; d = dst[eid]; w = ew[eid]; }
  else         { s = eid - E;  d = s;        w = 1.0f;    }
  float nrm = dis[s] * w * dis[d];
  float4 v = *(const float4*)(MLV + (size_t)s * HID_CH + j);
  float* o = (j < OUT_CH) ? (mu + (size_t)d * OUT_CH + j)
                          : (lv + (size_t)d * OUT_CH + (j - OUT_CH));
  atomicAdd(o + 0, v.x * nrm);
  atomicAdd(o + 1, v.y * nrm);
  atomicAdd(o + 2, v.z * nrm);
  atomicAdd(o + 3, v.w * nrm);
}

__global__ void bias2_kernel(float* __restrict__ out, const float* __restrict__ bmu,
                             const float* __restrict__ blv, long long half_total) {
  long long i = (long long)blockIdx.x * blockDim.x + threadIdx.x;
  long long stride = (long long)gridDim.x * blockDim.x;
  long long total = 2 * half_total;
  for (; i < total; i += stride) {
    int col = (int)(i & (OUT_CH - 1));
    out[i] += (i < half_total) ? bmu[col] : blv[col];
  }
}

// ---------------- launch ----------------
extern "C" void kernel_launch(void* const* d_in, const int* in_sizes, int n_in,
                              void* d_out, int out_size, void* d_ws, size_t ws_size,
                              hipStream_t stream) {
  const float* x   = (const float*)d_in[0];
  const int*   ei  = (const int*)d_in[1];
  const float* ew  = (const float*)d_in[2];
  const float* W1  = (const float*)d_in[3];
  const float* b1  = (const float*)d_in[4];
  const float* Wmu = (const float*)d_in[5];
  const float* bmu = (const float*)d_in[6];
  const float* Wlv = (const float*)d_in[7];
  const float* blv = (const float*)d_in[8];
  float* out = (float*)d_out;

  const int Nn = in_sizes[0] / IN_CH;   // 100000
  const int E  = in_sizes[2];           // 1600000
  const int* src = ei;                  // edge_index row 0
  const int* dst = ei + E;              // edge_index row 1

  float* ws   = (float*)d_ws;
  float* dis  = ws;                                  // [Nn]   degree -> rsqrt
  float* Hlin = ws + Nn;                             // [Nn*128] (reused for MLV pre-scatter)
  float* H    = Hlin + (size_t)Nn * HID_CH;          // [Nn*128]

  // init: deg starts at 1.0 (self-loop weight folded in); scatter targets zeroed
  fill_kernel<<<1024, 256, 0, stream>>>(dis, 1.0f, Nn);
  fill_kernel<<<4096, 256, 0, stream>>>(H, 0.0f, (long long)Nn * HID_CH);
  fill_kernel<<<4096, 256, 0, stream>>>(out, 0.0f, (long long)Nn * 2 * OUT_CH);

  // symmetric GCN normalization
  deg_kernel<<<(E + 255) / 256, 256, 0, stream>>>(dst, ew, dis, E);
  dinv_kernel<<<(Nn + 255) / 256, 256, 0, stream>>>(dis, Nn);

  // layer 1
  int mtiles = Nn / 16;                 // 100000 / 16 = 6250, exact
  gemm1_kernel<<<(mtiles + 7) / 8, 256, 0, stream>>>(x, W1, Hlin, mtiles);
  int Etot = E + Nn;
  scatter1_kernel<<<(Etot + 7) / 8, 256, 0, stream>>>(src, dst, ew, dis, Hlin, H, E, Nn);
  bias_relu_kernel<<<4096, 256, 0, stream>>>(H, b1, (long long)Nn * HID_CH);

  // layer 2 (mu | logvar fused in one 128-wide pass)
  gemm2_kernel<<<(mtiles + 7) / 8, 256, 0, stream>>>(H, Wmu, Wlv, Hlin, mtiles);
  scatter2_kernel<<<(Etot + 7) / 8, 256, 0, stream>>>(src, dst, ew, dis, Hlin,
                                                      out, out + (size_t)Nn * OUT_CH, E, Nn);
  bias2_kernel<<<4096, 256, 0, stream>>>(out, bmu, blv, (long long)Nn * OUT_CH);
}